// MoEGate_22436909154531
// MI455X (gfx1250) — compile-verified
//
#include <hip/hip_runtime.h>

// MoE gating for MI455X (gfx1250, wave32):
//   logits = x @ W_g^T ([32768,4096] x [4096,64]) via V_WMMA_F32_16X16X4_F32,
//   W_g staged through LDS with triple-buffered async-to-LDS copies,
//   then per-row softmax over 64 experts and top-8 (vals f32, idx i32).
//
// Roofline: 17.2 GFLOP vs 512 MB of x traffic -> ~22 us HBM floor @ 23.3 TB/s.
// W_g (1 MB) is L2-resident; LDS staging cuts per-wave B refetch 8x per WG and
// moves B-operand bandwidth from the vmem pipe onto the LDS pipe.

typedef __attribute__((ext_vector_type(2))) float v2f;
typedef __attribute__((ext_vector_type(4))) float v4f;
typedef __attribute__((ext_vector_type(8))) float v8f;
typedef __attribute__((ext_vector_type(4))) int   v4i;

// Builtin wants (AS1 v4i*, AS3 v4i*, imm offset, imm cpol) per hipcc diagnostic.
typedef __attribute__((address_space(1))) v4i* gbl_v4i_p;
typedef __attribute__((address_space(3))) v4i* lds_v4i_p;

#define D_HIDDEN     4096
#define N_EXPERTS    64
#define TOPK         8
#define KC           64                       // K-chunk of W_g staged in LDS
#define CSTRIDE      (KC + 2)                 // 66: bank = (2e + k) % 64, conflict-free b64 reads
#define CHUNK_FLOATS (N_EXPERTS * CSTRIDE)    // 4224 floats = 16.5 KB
#define NBUF         3                        // prefetch depth 2
#define NCHUNKS      (D_HIDDEN / KC)          // 64
#define WAVES_PER_WG 8
#define ROWS_PER_WG  128                      // 8 waves * 16 rows

#if __has_builtin(__builtin_amdgcn_global_load_async_to_lds_b128)
#define HAVE_ASYNC 1
#else
#define HAVE_ASYNC 0
#endif

#if __has_builtin(__builtin_amdgcn_s_wait_asynccnt)
#define WAIT_ASYNC(n) __builtin_amdgcn_s_wait_asynccnt(n)
#else
#define WAIT_ASYNC(n) asm volatile("s_wait_asynccnt %0" ::"n"(n))
#endif

__global__ __launch_bounds__(256, 1)
void moe_gate_wmma(const float* __restrict__ x,
                   const float* __restrict__ wg,
                   float* __restrict__ vals_out,
                   int*   __restrict__ idx_out)
{
    // 3 chunk buffers; same region reused for the 8x(16x66) epilogue tiles.
    __shared__ float smem[NBUF * CHUNK_FLOATS];   // 50,688 B

    const int tid  = threadIdx.x;
    const int lane = tid & 31;
    const int wave = tid >> 5;
    const int lmod = lane & 15;   // M (A) / N (B) / column (C)
    const int lgrp = lane >> 4;   // K-phase: lanes 0-15 -> K={0,1}, 16-31 -> K={2,3}

    const int row0 = blockIdx.x * ROWS_PER_WG + wave * 16;

    // A operand: lane l, VGPR v -> x[row0 + l%16][k + (l>>4)*2 + v]
    const float* ap = x + (size_t)(row0 + lmod) * D_HIDDEN + (lgrp << 1);

    // Stage one K-chunk of W_g (all 64 experts x KC floats) into LDS buffer c%3.
    // 1024 b128 slots, 4 per thread; coalesced 16B chunks along each expert row.
    auto stage = [&](int c) {
        const int p = c % NBUF;
        #pragma unroll
        for (int i = 0; i < 4; ++i) {
            const int s  = i * 256 + tid;     // slot id
            const int e  = s >> 4;            // expert (16 slots per row)
            const int c4 = (s & 15) << 2;     // float column within chunk
            float* gsrc = (float*)wg + (size_t)e * D_HIDDEN + c * KC + c4;
            const int ld = p * CHUNK_FLOATS + e * CSTRIDE + c4;
#if HAVE_ASYNC
            __builtin_amdgcn_global_load_async_to_lds_b128(
                (gbl_v4i_p)(v4i*)gsrc,
                (lds_v4i_p)(v4i*)&smem[ld],
                0, 0);
#else
            *(v4f*)&smem[ld] = *(const v4f*)gsrc;
#endif
        }
    };

    // B operand (LDS floats, within a buffer): expert n = t*16 + l%16,
    // lane l, VGPR v -> W_chunk[n][kk + (l>>4)*2 + v]
    const int boff0 = (lmod +  0) * CSTRIDE + (lgrp << 1);
    const int boff1 = (lmod + 16) * CSTRIDE + (lgrp << 1);
    const int boff2 = (lmod + 32) * CSTRIDE + (lgrp << 1);
    const int boff3 = (lmod + 48) * CSTRIDE + (lgrp << 1);

    v8f acc0 = {0.f,0.f,0.f,0.f,0.f,0.f,0.f,0.f};
    v8f acc1 = acc0, acc2 = acc0, acc3 = acc0;

    // Prologue: two chunks in flight, wait for the first.
    stage(0);
    stage(1);
    WAIT_ASYNC(4);
    __syncthreads();

    for (int c = 0; c < NCHUNKS; ++c) {
        if (c + 2 < NCHUNKS) stage(c + 2);

        const int bo = (c % NBUF) * CHUNK_FLOATS;
        const float* apc = ap + c * KC;

        #pragma unroll 8
        for (int kk = 0; kk < KC; kk += 4) {
            v2f a  = *(const v2f*)(apc + kk);
            v2f b0 = *(const v2f*)&smem[bo + boff0 + kk];
            v2f b1 = *(const v2f*)&smem[bo + boff1 + kk];
            v2f b2 = *(const v2f*)&smem[bo + boff2 + kk];
            v2f b3 = *(const v2f*)&smem[bo + boff3 + kk];
            // 8 args: (neg_a, A, neg_b, B, c_mod, C, reuse_a, reuse_b)
            acc0 = __builtin_amdgcn_wmma_f32_16x16x4_f32(false, a, false, b0, (short)0, acc0, false, false);
            acc1 = __builtin_amdgcn_wmma_f32_16x16x4_f32(false, a, false, b1, (short)0, acc1, false, false);
            acc2 = __builtin_amdgcn_wmma_f32_16x16x4_f32(false, a, false, b2, (short)0, acc2, false, false);
            acc3 = __builtin_amdgcn_wmma_f32_16x16x4_f32(false, a, false, b3, (short)0, acc3, false, false);
        }

        // In-order async completion: <=4 outstanding means chunk c+1 has landed.
        if (c + 2 < NCHUNKS) { WAIT_ASYNC(4); } else { WAIT_ASYNC(0); }
        __syncthreads();
    }

    // Epilogue: C/D layout is VGPR r, lanes 0-15 -> (M=r, N=lane),
    // lanes 16-31 -> (M=r+8, N=lane-16). Reuse smem (stride 66: bank =
    // (2*row + col) % 64, conflict-free for 16 concurrent row readers).
    float* my = smem + wave * 16 * CSTRIDE;
    #pragma unroll
    for (int r = 0; r < 8; ++r) {
        const int m = (lgrp << 3) + r;
        my[m * CSTRIDE +  0 + lmod] = acc0[r];
        my[m * CSTRIDE + 16 + lmod] = acc1[r];
        my[m * CSTRIDE + 32 + lmod] = acc2[r];
        my[m * CSTRIDE + 48 + lmod] = acc3[r];
    }
    // Same-wave LDS ops are in-order; each wave reads only its own slice.

    if (lane < 16) {
        const float* lrow = my + lane * CSTRIDE;

        float logit[N_EXPERTS];   // constant-indexed only -> stays in VGPRs
        float mx = -3.402823466e+38f;
        #pragma unroll
        for (int n = 0; n < N_EXPERTS; ++n) {
            logit[n] = lrow[n];
            mx = fmaxf(mx, logit[n]);
        }

        float denom = 0.f;
        #pragma unroll
        for (int n = 0; n < N_EXPERTS; ++n)
            denom += __expf(logit[n] - mx);
        const float inv = 1.f / denom;

        const size_t row = (size_t)row0 + lane;
        unsigned long long taken = 0ull;
        #pragma unroll
        for (int j = 0; j < TOPK; ++j) {
            float best = -3.402823466e+38f;
            int   bi   = 0;
            #pragma unroll
            for (int n = 0; n < N_EXPERTS; ++n) {
                const bool avail = ((taken >> n) & 1ull) == 0ull;
                if (avail && (logit[n] > best)) { best = logit[n]; bi = n; }
            }
            taken |= (1ull << bi);
            vals_out[row * TOPK + j] = __expf(best - mx) * inv;
            idx_out [row * TOPK + j] = bi;
        }
    }
}

extern "C" void kernel_launch(void* const* d_in, const int* in_sizes, int n_in,
                              void* d_out, int out_size, void* d_ws, size_t ws_size,
                              hipStream_t stream) {
    const float* x  = (const float*)d_in[0];
    const float* wg = (const float*)d_in[1];
    const int n_tokens = in_sizes[0] / D_HIDDEN;   // 32768

    // d_out = [vals: n_tokens*8 f32][idx: n_tokens*8 i32], flat in return order.
    float* vals = (float*)d_out;
    int*   idx  = (int*)((float*)d_out + (size_t)n_tokens * TOPK);

    dim3 grid(n_tokens / ROWS_PER_WG);   // 256 workgroups
    dim3 block(256);                     // 8 wave32s
    moe_gate_wmma<<<grid, block, 0, stream>>>(x, wg, vals, idx);
}